// PaiNNVelocityNetwork_88897233093050
// MI455X (gfx1250) — compile-verified
//
#include <hip/hip_runtime.h>
#include <math.h>

// Problem constants (match reference)
#define BB 4
#define NN 256
#define HH 128
#define LL 5
#define NRBF 20
#define BN (BB*NN)
#define CUT 10.0f
#define JSPLIT 4
#define MSG_THREADS 512
#define PI_F 3.14159265358979323846f

typedef __attribute__((ext_vector_type(2))) float v2f;
typedef __attribute__((ext_vector_type(8))) float v8f;

__device__ __forceinline__ float silu_f(float x){ return x / (1.f + __expf(-x)); }

// ---------------------------------------------------------------------------
// Geometry precompute: dist -> dir, rbf(20), f_cut (diagonal masked to 0)
// ---------------------------------------------------------------------------
__global__ void geom_kernel(const float* __restrict__ pos, float* __restrict__ rbf,
                            float* __restrict__ fcut, float* __restrict__ dir)
{
  int idx = blockIdx.x*blockDim.x + threadIdx.x;
  if (idx >= BB*NN*NN) return;
  int j = idx % NN; int i = (idx/NN) % NN; int b = idx/(NN*NN);
  const float* pi = pos + (long)(b*NN+i)*3;
  const float* pj = pos + (long)(b*NN+j)*3;
  float dx = pj[0]-pi[0], dy = pj[1]-pi[1], dz = pj[2]-pi[2];
  float dist = sqrtf(dx*dx + dy*dy + dz*dz);
  float inv  = 1.f/(dist + 1e-8f);
  dir[(long)idx*3+0] = dx*inv;
  dir[(long)idx*3+1] = dy*inv;
  dir[(long)idx*3+2] = dz*inv;
  const float width = CUT/(NRBF-1);
  #pragma unroll
  for (int k=0;k<NRBF;k++){
    float u = (dist - (float)k*width)/width;
    rbf[(long)idx*NRBF + k] = __expf(-0.5f*u*u);
  }
  float fc = 0.f;
  if (i != j && dist < CUT) fc = 0.5f*(1.f + __cosf(PI_F*dist/CUT));
  fcut[idx] = fc;
}

// ---------------------------------------------------------------------------
// Time embedding + projection
// ---------------------------------------------------------------------------
__global__ void temb_kernel(const float* __restrict__ t, const float* __restrict__ W,
                            const float* __restrict__ bv, float* __restrict__ tproj)
{
  __shared__ float te[HH];
  int b = blockIdx.x, h = threadIdx.x;
  float tv = t[b];
  int k = (h < 64) ? h : h - 64;
  float f = __expf(-logf(10000.f) * (float)k / 64.f);
  te[h] = (h < 64) ? __sinf(tv*f) : __cosf(tv*f);
  __syncthreads();
  float acc = bv[h];
  for (int i=0;i<HH;i++) acc += te[i]*W[i*HH + h];
  tproj[b*HH + h] = acc;
}

__global__ void sinit_kernel(const float* __restrict__ ae, const float* __restrict__ te,
                             const int* __restrict__ ty, const float* __restrict__ tp,
                             float* __restrict__ s)
{
  int idx = blockIdx.x*blockDim.x + threadIdx.x;
  if (idx >= BN*HH) return;
  int node = idx >> 7, h = idx & (HH-1);
  int b = node / NN, nn = node % NN;
  s[idx] = ae[h] + te[ty[nn]*HH + h] + tp[b*HH + h];
}

__global__ void vzero_kernel(float* __restrict__ v)
{
  int idx = blockIdx.x*blockDim.x + threadIdx.x;
  if (idx < BN*3*HH) v[idx] = 0.f;
}

// ---------------------------------------------------------------------------
// Weight transpose: out[n*K + k] = in[k*N + n]   (in: K x N, out: N x K)
// Makes each WMMA B-fragment a single contiguous b64 load.
// ---------------------------------------------------------------------------
__global__ void transpose_kernel(const float* __restrict__ in, float* __restrict__ out,
                                 int K, int Nc)
{
  int idx = blockIdx.x*blockDim.x + threadIdx.x;
  if (idx >= K*Nc) return;
  int k = idx / Nc, n = idx % Nc;
  out[(long)n*K + k] = in[idx];
}

// ---------------------------------------------------------------------------
// LayerNorm over H=128, writes row with configurable stride (for concat)
// ---------------------------------------------------------------------------
__global__ void ln_kernel(const float* __restrict__ x, const float* __restrict__ g,
                          const float* __restrict__ bt, float* __restrict__ out,
                          int ostride)
{
  __shared__ float red[HH];
  int row = blockIdx.x, h = threadIdx.x;
  float val = x[(long)row*HH + h];
  red[h] = val; __syncthreads();
  for (int s=64;s>0;s>>=1){ if (h<s) red[h]+=red[h+s]; __syncthreads(); }
  float mu = red[0]*(1.f/HH); __syncthreads();
  float d = val - mu;
  red[h] = d*d; __syncthreads();
  for (int s=64;s>0;s>>=1){ if (h<s) red[h]+=red[h+s]; __syncthreads(); }
  float var = red[0]*(1.f/HH);
  out[(long)row*ostride + h] = d*rsqrtf(var + 1e-5f)*g[h] + bt[h];
}

// ---------------------------------------------------------------------------
// Generic fp32 WMMA GEMM: C = act(A[MxK] @ B[KxN] + bias)
// B is supplied TRANSPOSED (BT: N x K). Each wave computes TWO M-tiles
// (32 rows x 16 cols) sharing one B fragment. Fragments for 64 K-values are
// preloaded into register arrays (batched b64 load clause) before the WMMA
// chain, so loads overlap the matrix pipe instead of serializing it.
// Requirements: K % 64 == 0, M % 32 == 0, N % 16 == 0, (M/32)*(N/16) % 8 == 0.
// ---------------------------------------------------------------------------
template<int ACT>
__global__ __launch_bounds__(256) void gemm_kernel(
    const float* __restrict__ A, const float* __restrict__ BT,
    const float* __restrict__ bias, float* __restrict__ C,
    int M, int Nc, int K)
{
  const int lane = threadIdx.x & 31;
  const int wave = threadIdx.x >> 5;
  const int ntn  = Nc >> 4;
  int g   = blockIdx.x*8 + wave;
  int mt2 = g / ntn, nt = g % ntn;
  if (mt2*32 >= M) return;
  const int row = lane & 15;       // A row within M-tile / B,C column lane
  const int kh  = lane >> 4;       // K-half (A/B frag) and row-half (C frag)
  const int col = nt*16 + row;
  float bv = bias ? bias[col] : 0.f;
  v8f accA = {bv,bv,bv,bv,bv,bv,bv,bv};
  v8f accB = accA;
  const float* ArA = A  + (long)(mt2*32 + row)*K;
  const float* ArB = ArA + (long)16*K;
  const float* Bc  = BT + (long)col*K;
  for (int kc = 0; kc < K; kc += 64){
    v2f bfr[16], a0r[16], a1r[16];
    #pragma unroll
    for (int s=0;s<16;s++){
      int ka = kc + 4*s + 2*kh;
      bfr[s] = *(const v2f*)(Bc  + ka);
      a0r[s] = *(const v2f*)(ArA + ka);
      a1r[s] = *(const v2f*)(ArB + ka);
    }
    #pragma unroll
    for (int s=0;s<16;s++){
      accA = __builtin_amdgcn_wmma_f32_16x16x4_f32(false, a0r[s], false, bfr[s],
                                                   (short)0, accA, false, false);
      accB = __builtin_amdgcn_wmma_f32_16x16x4_f32(false, a1r[s], false, bfr[s],
                                                   (short)0, accB, false, false);
    }
  }
  #pragma unroll
  for (int u=0;u<8;u++){
    int r = mt2*32 + u + 8*kh;
    float x = accA[u];
    if (ACT) x = silu_f(x);
    C[(long)r*Nc + col] = x;
  }
  #pragma unroll
  for (int u=0;u<8;u++){
    int r = mt2*32 + 16 + u + 8*kh;
    float x = accB[u];
    if (ACT) x = silu_f(x);
    C[(long)r*Nc + col] = x;
  }
}

// ---------------------------------------------------------------------------
// Fused message-passing kernel.
// grid = B * (N/16) * JSPLIT, block = 512 (16 waves). Each wave owns ONE i-row
// of the 16-wide i-tile. Per 16-wide j-tile: stage 1 builds filter hidden
// activations (K=20, VALU, weights staged in LDS) directly in WMMA A-fragment
// layout; stage 2 preloads all 32 B-fragments (W2 transposed, b64 clause)
// then runs the 32-WMMA chain, and fuses f_cut * ctx[j] * 1/sqrt(N-1),
// dir / v[j] weighting, j-reduction (in-register + shfl_xor 16) and LDS
// accumulation. f_cut and dir tiles are hoisted out of the nt loop.
// Per-j-slice partials go to dS/dV (deterministic, no atomics).
// ---------------------------------------------------------------------------
__global__ __launch_bounds__(MSG_THREADS) void msg_kernel(
    const float* __restrict__ rbf, const float* __restrict__ fcut,
    const float* __restrict__ dir, const float* __restrict__ ctx,
    const float* __restrict__ v,
    const float* __restrict__ W1, const float* __restrict__ b1,
    const float* __restrict__ W2T, const float* __restrict__ b2,
    float* __restrict__ dS, float* __restrict__ dV)
{
  __shared__ float sacc[16*HH];       //  8 KB
  __shared__ float vacc[16*3*HH];     // 24 KB
  __shared__ float w1t[HH*NRBF];      // 10 KB : W1 transposed [k][q]
  __shared__ float b1s[HH];
  const int tid = threadIdx.x;
  for (int x = tid; x < 16*HH;   x += MSG_THREADS) sacc[x] = 0.f;
  for (int x = tid; x < 16*3*HH; x += MSG_THREADS) vacc[x] = 0.f;
  for (int x = tid; x < HH*NRBF; x += MSG_THREADS){
    int k = x / NRBF, q = x % NRBF;
    w1t[x] = W1[q*HH + k];
  }
  if (tid < HH) b1s[tid] = b1[tid];
  __syncthreads();

  const int wg = blockIdx.x;
  const int js = wg % JSPLIT;
  const int it = (wg / JSPLIT) % (NN/16);
  const int b  = wg / (JSPLIT * (NN/16));
  const int i0 = it*16;
  const int lane = tid & 31;
  const int wave = tid >> 5;        // 0..15 == M-tile == i row
  const int eRow = lane & 15;       // j row within tile (A-matrix row)
  const int kh   = lane >> 4;       // K-half / row-half
  const int colSub = lane & 15;     // output feature lane
  const float INVS = rsqrtf((float)(NN-1));

  const int mt = wave;
  const int i  = i0 + mt;
  const long irow = (long)(b*NN + i)*NN;

  for (int jt = 0; jt < (NN/16)/JSPLIT; ++jt) {
    const int j0 = js*(NN/JSPLIT) + jt*16;
    const long ebase = irow + (j0 + eRow);
    if (jt + 1 < (NN/16)/JSPLIT)
      __builtin_prefetch(&rbf[(ebase + 16)*NRBF], 0, 3);   // next j-tile rbf

    // ---- stage 1: hidden = silu(rbf @ W1 + b1), built as A-fragments ----
    float r[NRBF];
    {
      const float4* rp = (const float4*)&rbf[ebase*NRBF];  // 80B-aligned row
      #pragma unroll
      for (int q=0;q<NRBF/4;q++){
        float4 rv = rp[q];
        r[q*4+0] = rv.x; r[q*4+1] = rv.y; r[q*4+2] = rv.z; r[q*4+3] = rv.w;
      }
    }
    v2f afr[32];
    #pragma unroll
    for (int s=0;s<32;s++){
      int k0 = 4*s + 2*kh;
      const float* wr0 = &w1t[k0*NRBF];
      const float* wr1 = &w1t[(k0+1)*NRBF];
      float h0 = b1s[k0], h1 = b1s[k0+1];
      #pragma unroll
      for (int q=0;q<NRBF;q++){
        h0 += r[q]*wr0[q];
        h1 += r[q]*wr1[q];
      }
      afr[s].x = silu_f(h0);
      afr[s].y = silu_f(h1);
    }

    // ---- hoisted per-j-tile edge scalars (invariant over nt) ----
    float fc[8];                 // f_cut / sqrt(N-1) for this lane's 8 j-rows
    #pragma unroll
    for (int u=0;u<8;u++) fc[u] = fcut[irow + j0 + u + 8*kh] * INVS;
    float dirv[3][8];
    #pragma unroll
    for (int d=0;d<3;d++)
      #pragma unroll
      for (int u=0;u<8;u++) dirv[d][u] = dir[(irow + j0 + u + 8*kh)*3 + d];

    // ---- stage 2: Wf = hidden @ W2 + b2 (WMMA), fused post-processing ----
    for (int nt = 0; nt < 24; ++nt) {
      const int n = nt*16 + colSub;
      const float* Bc = W2T + (long)n*HH;
      v2f bfr[32];
      #pragma unroll
      for (int s=0;s<32;s++)
        bfr[s] = *(const v2f*)(Bc + 4*s + 2*kh);
      float bv = b2[n];
      v8f acc = {bv,bv,bv,bv,bv,bv,bv,bv};
      #pragma unroll
      for (int s=0;s<32;s++)
        acc = __builtin_amdgcn_wmma_f32_16x16x4_f32(false, afr[s], false, bfr[s],
                                                    (short)0, acc, false, false);
      // msg = Wf * f_cut * ctx[j] / sqrt(N-1); D-frag row u -> j = j0+u+8*kh
      float val[8];
      #pragma unroll
      for (int u=0;u<8;u++){
        int jj = j0 + u + 8*kh;
        float cx = ctx[(long)(b*NN+jj)*(3*HH) + n];
        val[u] = acc[u]*fc[u]*cx;
      }
      if (nt < 8) {                       // ds channel -> s accumulation
        float sum = 0.f;
        #pragma unroll
        for (int u=0;u<8;u++) sum += val[u];
        sum += __shfl_xor(sum, 16, 32);
        if (kh == 0) sacc[mt*HH + n] += sum;
      } else if (nt < 16) {               // dvs channel -> dir-weighted
        int h = n - HH;
        #pragma unroll
        for (int d=0;d<3;d++){
          float sd = 0.f;
          #pragma unroll
          for (int u=0;u<8;u++) sd += val[u]*dirv[d][u];
          sd += __shfl_xor(sd, 16, 32);
          if (kh == 0) vacc[(mt*3 + d)*HH + h] += sd;
        }
      } else {                            // dvv channel -> v[j]-weighted
        int h = n - 2*HH;
        #pragma unroll
        for (int d=0;d<3;d++){
          float sd = 0.f;
          #pragma unroll
          for (int u=0;u<8;u++){
            int jj = j0 + u + 8*kh;
            sd += val[u]*v[((long)(b*NN+jj)*3 + d)*HH + h];
          }
          sd += __shfl_xor(sd, 16, 32);
          if (kh == 0) vacc[(mt*3 + d)*HH + h] += sd;
        }
      }
    }
  }
  __syncthreads();
  for (int x = tid; x < 16*HH; x += MSG_THREADS){
    int ii = x >> 7; int h = x & (HH-1);
    dS[(long)js*BN*HH + (long)(b*NN + i0 + ii)*HH + h] = sacc[x];
  }
  for (int x = tid; x < 16*3*HH; x += MSG_THREADS){
    int ii = x / (3*HH); int rest = x % (3*HH);
    dV[(long)js*BN*3*HH + (long)(b*NN + i0 + ii)*3*HH + rest] = vacc[x];
  }
}

// Reduce j-slice partials and apply: s += sum_js dS ; v += sum_js dV
__global__ void apply_kernel(float* __restrict__ s, float* __restrict__ v,
                             const float* __restrict__ dS, const float* __restrict__ dV)
{
  int idx = blockIdx.x*blockDim.x + threadIdx.x;
  if (idx >= BN*3*HH) return;
  float a = 0.f;
  #pragma unroll
  for (int js=0;js<JSPLIT;js++) a += dV[(long)js*BN*3*HH + idx];
  v[idx] += a;
  if (idx < BN*HH){
    float c = 0.f;
    #pragma unroll
    for (int js=0;js<JSPLIT;js++) c += dS[(long)js*BN*HH + idx];
    s[idx] += c;
  }
}

// Vv_norm -> cat[:,128:256]; dot_uv = sum_d Uv*Vv
__global__ void vnorm_dot_kernel(const float* __restrict__ Uv, const float* __restrict__ Vv,
                                 float* __restrict__ catb, float* __restrict__ dotuv)
{
  int idx = blockIdx.x*blockDim.x + threadIdx.x;
  if (idx >= BN*HH) return;
  int node = idx >> 7, h = idx & (HH-1);
  long base = (long)node*3*HH + h;
  float u0=Uv[base], u1=Uv[base+HH], u2=Uv[base+2*HH];
  float w0=Vv[base], w1=Vv[base+HH], w2=Vv[base+2*HH];
  catb[(long)node*(2*HH) + HH + h] = sqrtf(w0*w0 + w1*w1 + w2*w2 + 1e-8f);
  dotuv[idx] = u0*w0 + u1*w1 + u2*w2;
}

// s += a_ss + a_sv*dot ; v += a_vv * Uv
__global__ void update_kernel(float* __restrict__ s, float* __restrict__ v,
                              const float* __restrict__ ctx2,
                              const float* __restrict__ dotuv,
                              const float* __restrict__ Uv)
{
  int idx = blockIdx.x*blockDim.x + threadIdx.x;
  if (idx >= BN*HH) return;
  int node = idx >> 7, h = idx & (HH-1);
  long cb = (long)node*3*HH;
  float a_ss = ctx2[cb + h];
  float a_sv = ctx2[cb + HH + h];
  float a_vv = ctx2[cb + 2*HH + h];
  s[idx] += a_ss + a_sv*dotuv[idx];
  #pragma unroll
  for (int d=0;d<3;d++){
    long vi = cb + (long)d*HH + h;
    v[vi] += a_vv*Uv[vi];
  }
}

// velocity[node,d] = sum_h v[node,d,h]*rw[h]
__global__ void readout_kernel(const float* __restrict__ v, const float* __restrict__ rw,
                               float* __restrict__ out)
{
  __shared__ float red[HH];
  int node = blockIdx.x, h = threadIdx.x;
  float w = rw[h];
  for (int d=0; d<3; d++){
    red[h] = v[(long)node*3*HH + (long)d*HH + h]*w;
    __syncthreads();
    for (int s=64;s>0;s>>=1){ if (h<s) red[h]+=red[h+s]; __syncthreads(); }
    if (h==0) out[node*3 + d] = red[0];
    __syncthreads();
  }
}

// ---------------------------------------------------------------------------
extern "C" void kernel_launch(void* const* d_in, const int* in_sizes, int n_in,
                              void* d_out, int out_size, void* d_ws, size_t ws_size,
                              hipStream_t stream)
{
  (void)in_sizes; (void)n_in; (void)out_size; (void)ws_size;
  const float* pos      = (const float*)d_in[0];
  const float* t        = (const float*)d_in[1];
  const int*   ty       = (const int*)  d_in[2];
  const float* atom_emb = (const float*)d_in[3];
  const float* type_emb = (const float*)d_in[4];
  const float* tp_w     = (const float*)d_in[5];
  const float* tp_b     = (const float*)d_in[6];
  const float* int_ln_g = (const float*)d_in[7];
  const float* int_ln_b = (const float*)d_in[8];
  const float* ctx_w1   = (const float*)d_in[9];
  const float* ctx_b1   = (const float*)d_in[10];
  const float* ctx_w2   = (const float*)d_in[11];
  const float* ctx_b2   = (const float*)d_in[12];
  const float* filt_w1  = (const float*)d_in[13];
  const float* filt_b1  = (const float*)d_in[14];
  const float* filt_w2  = (const float*)d_in[15];
  const float* filt_b2  = (const float*)d_in[16];
  const float* mix_ln_g = (const float*)d_in[17];
  const float* mix_ln_b = (const float*)d_in[18];
  const float* Uw       = (const float*)d_in[19];
  const float* Vw       = (const float*)d_in[20];
  const float* mix_w1   = (const float*)d_in[21];
  const float* mix_b1   = (const float*)d_in[22];
  const float* mix_w2   = (const float*)d_in[23];
  const float* mix_b2   = (const float*)d_in[24];
  // d_in[25]/[26]: final_ln_g/b — unused by the reference forward
  const float* rw       = (const float*)d_in[27];

  float* W = (float*)d_ws;
  float* s     = W; W += BN*HH;
  float* v     = W; W += BN*3*HH;
  float* sn    = W; W += BN*HH;
  float* h1    = W; W += BN*HH;
  float* ctxn  = W; W += BN*3*HH;
  float* dS    = W; W += (size_t)JSPLIT*BN*HH;
  float* dV    = W; W += (size_t)JSPLIT*BN*3*HH;
  float* Uv    = W; W += BN*3*HH;
  float* Vv    = W; W += BN*3*HH;
  float* catb  = W; W += BN*2*HH;
  float* dotuv = W; W += BN*HH;
  float* hmix  = W; W += BN*HH;
  float* ctx2  = W; W += BN*3*HH;
  float* tproj = W; W += BB*HH;
  float* wT    = W; W += (size_t)3*HH*HH;     // transposed-weight scratch (max 128x384)
  float* rbf   = W; W += (size_t)BB*NN*NN*NRBF;
  float* fcut  = W; W += (size_t)BB*NN*NN;
  float* dir   = W; W += (size_t)BB*NN*NN*3;

  dim3 blk(256);
  dim3 mblk(MSG_THREADS);
  geom_kernel <<<dim3((BB*NN*NN + 255)/256), blk, 0, stream>>>(pos, rbf, fcut, dir);
  temb_kernel <<<dim3(BB), dim3(HH), 0, stream>>>(t, tp_w, tp_b, tproj);
  sinit_kernel<<<dim3((BN*HH + 255)/256), blk, 0, stream>>>(atom_emb, type_emb, ty, tproj, s);
  vzero_kernel<<<dim3((BN*3*HH + 255)/256), blk, 0, stream>>>(v);

  #define TRANS(src, K_, N_) \
    transpose_kernel<<<dim3(((K_)*(N_) + 255)/256), blk, 0, stream>>>((src), wT, (K_), (N_))

  for (int l = 0; l < LL; ++l){
    ln_kernel<<<dim3(BN), dim3(HH), 0, stream>>>(s, int_ln_g + l*HH, int_ln_b + l*HH, sn, HH);

    TRANS(ctx_w1 + (long)l*HH*HH, HH, HH);
    gemm_kernel<1><<<dim3((BN/32)*(HH/16)/8),   blk, 0, stream>>>(sn,  wT, ctx_b1 + l*HH,   h1,   BN, HH,   HH);
    TRANS(ctx_w2 + (long)l*HH*3*HH, HH, 3*HH);
    gemm_kernel<0><<<dim3((BN/32)*(3*HH/16)/8), blk, 0, stream>>>(h1,  wT, ctx_b2 + l*3*HH, ctxn, BN, 3*HH, HH);

    TRANS(filt_w2 + (long)l*HH*3*HH, HH, 3*HH);
    msg_kernel<<<dim3(BB*(NN/16)*JSPLIT), mblk, 0, stream>>>(rbf, fcut, dir, ctxn, v,
        filt_w1 + (long)l*NRBF*HH, filt_b1 + l*HH, wT, filt_b2 + l*3*HH, dS, dV);
    apply_kernel<<<dim3((BN*3*HH + 255)/256), blk, 0, stream>>>(s, v, dS, dV);

    TRANS(Uw + (long)l*HH*HH, HH, HH);
    gemm_kernel<0><<<dim3((BN*3/32)*(HH/16)/8), blk, 0, stream>>>(v, wT, nullptr, Uv, BN*3, HH, HH);
    TRANS(Vw + (long)l*HH*HH, HH, HH);
    gemm_kernel<0><<<dim3((BN*3/32)*(HH/16)/8), blk, 0, stream>>>(v, wT, nullptr, Vv, BN*3, HH, HH);

    vnorm_dot_kernel<<<dim3((BN*HH + 255)/256), blk, 0, stream>>>(Uv, Vv, catb, dotuv);
    ln_kernel<<<dim3(BN), dim3(HH), 0, stream>>>(s, mix_ln_g + l*HH, mix_ln_b + l*HH, catb, 2*HH);

    TRANS(mix_w1 + (long)l*2*HH*HH, 2*HH, HH);
    gemm_kernel<1><<<dim3((BN/32)*(HH/16)/8),   blk, 0, stream>>>(catb, wT, mix_b1 + l*HH,   hmix, BN, HH,   2*HH);
    TRANS(mix_w2 + (long)l*HH*3*HH, HH, 3*HH);
    gemm_kernel<0><<<dim3((BN/32)*(3*HH/16)/8), blk, 0, stream>>>(hmix, wT, mix_b2 + l*3*HH, ctx2, BN, 3*HH, HH);

    update_kernel<<<dim3((BN*HH + 255)/256), blk, 0, stream>>>(s, v, ctx2, dotuv, Uv);
  }
  #undef TRANS

  readout_kernel<<<dim3(BN), dim3(HH), 0, stream>>>(v, rw, (float*)d_out);
}